// EdgeConv_76398878261700
// MI455X (gfx1250) — compile-verified
//
#include <hip/hip_runtime.h>

#define B_    8
#define C_    64
#define N_    4096
#define CO_   64
#define K_    20
#define KP1_  21
#define BN_EPS 1e-5f
#define SLOPE  0.2f
#define CHUNK  1024

typedef float v2f __attribute__((ext_vector_type(2)));
typedef float v8f __attribute__((ext_vector_type(8)));

// ---------------------------------------------------------------- sq norms
__global__ void sq_kernel(const float* __restrict__ x, float* __restrict__ sq) {
    int i = blockIdx.x * blockDim.x + threadIdx.x;           // 0 .. B*N-1
    if (i >= B_ * N_) return;
    int b = i >> 12, n = i & (N_ - 1);
    const float* xb = x + (size_t)b * C_ * N_;
    float a = xb[n], c = xb[N_ + n], d = xb[2 * N_ + n];
    sq[i] = a * a + c * c + d * d;
}

// ---------------------------------------------------------------- kNN (top-21, drop self)
// Candidates staged in LDS as packed float4 (x,y,z,|p|^2): one broadcast
// ds_load_b128 per candidate in the scan loop.
__global__ void knn_kernel(const float* __restrict__ x, const float* __restrict__ sq,
                           int* __restrict__ edges) {
    __shared__ float4 spt[CHUNK];                            // 16 KB
    const int tiles = N_ / 256;
    int b = blockIdx.x / tiles;
    int n = (blockIdx.x % tiles) * 256 + threadIdx.x;
    const float* xb = x + (size_t)b * C_ * N_;
    float px = xb[n], py = xb[N_ + n], pz = xb[2 * N_ + n];
    float msq = sq[b * N_ + n];

    float bd[KP1_]; int bi[KP1_];
#pragma unroll
    for (int j = 0; j < KP1_; ++j) { bd[j] = 3.0e38f; bi[j] = -1; }

    for (int m0 = 0; m0 < N_; m0 += CHUNK) {
        __syncthreads();
        for (int j = threadIdx.x; j < CHUNK; j += 256) {
            float4 v;
            v.x = xb[m0 + j];
            v.y = xb[N_ + m0 + j];
            v.z = xb[2 * N_ + m0 + j];
            v.w = sq[b * N_ + m0 + j];
            spt[j] = v;
        }
        __syncthreads();
        for (int j = 0; j < CHUNK; ++j) {
            float4 q = spt[j];                               // broadcast: conflict-free
            float d = msq + q.w - 2.0f * (px * q.x + py * q.y + pz * q.z);
            if (d < bd[KP1_ - 1]) {                 // strict: keeps lowest-index ties first
                int pos = KP1_ - 1;
                while (pos > 0 && bd[pos - 1] > d) {
                    bd[pos] = bd[pos - 1]; bi[pos] = bi[pos - 1]; --pos;
                }
                bd[pos] = d; bi[pos] = m0 + j;
            }
        }
    }
    int* e = edges + ((size_t)b * N_ + n) * K_;
    int w = 0;
    for (int j = 0; j < KP1_ && w < K_; ++j) {
        if (bi[j] == n) continue;                   // drop self
        e[w++] = bi[j];
    }
}

// ---------------------------------------------------------------- Wcat^T prep: [W1-W2 ; W2]
__global__ void wcat_kernel(const float* __restrict__ W, float* __restrict__ wcatT) {
    int i = blockIdx.x * 256 + threadIdx.x;         // 64*128 entries
    if (i >= C_ * 128) return;
    int c = i >> 7, j = i & 127;                    // wcatT[c][j], j = output column of P
    float v = (j < CO_) ? (W[j * 128 + c] - W[j * 128 + 64 + c])
                        : W[(j - 64) * 128 + 64 + c];
    wcatT[c * 128 + j] = v;
}

// ---------------------------------------------------------------- projection GEMM (WMMA f32)
// P (B*N x 128) = xn (B*N x 64) @ wcatT (64 x 128).
// Block = 128 rows x 128 cols: sA (32KB) + sB (32KB) staged once via float4
// loads; each of 8 waves owns one 16-wide column stripe, keeps all 16
// B-fragments (32 VGPRs) in registers, sweeps 8 row sub-tiles -> 128 WMMAs.
#define GROWS 128
__global__ void proj_gemm_kernel(const float* __restrict__ x,
                                 const float* __restrict__ wcatT,
                                 float* __restrict__ P) {
    __shared__ float sA[C_ * GROWS];                // [c][r], 32 KB
    __shared__ float sB[C_ * 128];                  // [c][j], 32 KB
    int wave = threadIdx.x >> 5;
    int lane = threadIdx.x & 31;
    int m0 = blockIdx.x * GROWS;                    // 128 consecutive points, same batch
    int b  = m0 >> 12;
    int nbase = m0 & (N_ - 1);
    const float* xb = x + (size_t)b * C_ * N_;

    // stage B (float4, coalesced)
    {
        const float4* w4 = (const float4*)wcatT;
        float4* s4 = (float4*)sB;
        for (int i = threadIdx.x; i < C_ * 128 / 4; i += 256) s4[i] = w4[i];
    }
    // stage A: sA[c][r] = x[b][c][nbase+r]; rows of 128 are contiguous in x -> float4
    {
        float4* s4 = (float4*)sA;
        for (int i = threadIdx.x; i < C_ * GROWS / 4; i += 256) {
            int c = i >> 5, r4 = i & 31;            // 32 float4 per channel row
            s4[i] = ((const float4*)(xb + (size_t)c * N_ + nbase))[r4];
        }
    }
    __syncthreads();

    int j0  = wave * 16;
    int r   = lane & 15;
    int khi = (lane >> 4) << 1;                     // lanes 16-31 hold K+2 (ISA 16x4 f32 layout)

    // B fragments for the whole K range live in registers across row tiles
    v2f bfrag[16];
#pragma unroll
    for (int kk = 0; kk < 16; ++kk) {
        int c0 = kk * 4 + khi;
        bfrag[kk].x = sB[c0 * 128 + j0 + r];
        bfrag[kk].y = sB[(c0 + 1) * 128 + j0 + r];
    }

    for (int rt = 0; rt < GROWS / 16; ++rt) {
        int rowoff = rt * 16;
        v8f acc = {0.f, 0.f, 0.f, 0.f, 0.f, 0.f, 0.f, 0.f};
#pragma unroll
        for (int kk = 0; kk < 16; ++kk) {
            int c0 = kk * 4 + khi;
            v2f a;
            a.x = sA[c0 * GROWS + rowoff + r];
            a.y = sA[(c0 + 1) * GROWS + rowoff + r];
            acc = __builtin_amdgcn_wmma_f32_16x16x4_f32(
                      false, a, false, bfrag[kk], (short)0, acc, false, false);
        }
        int col  = j0 + r;
        int rowb = m0 + rowoff + ((lane >> 4) << 3);
#pragma unroll
        for (int t = 0; t < 8; ++t)
            P[(size_t)(rowb + t) * 128 + col] = acc[t];
    }
}

// ---------------------------------------------------------------- stats zero
__global__ void init_stats_kernel(float* s) {
    if (threadIdx.x < 128) s[threadIdx.x] = 0.f;    // gsum[64] + gsq[64] contiguous
}

// ---------------------------------------------------------------- per-channel sum / sumsq
__global__ void edge_stats_kernel(const float* __restrict__ P, const int* __restrict__ edges,
                                  float* __restrict__ gsum, float* __restrict__ gsq) {
    __shared__ float ssum[CO_], ssq2[CO_];
    if (threadIdx.x < CO_) { ssum[threadIdx.x] = 0.f; ssq2[threadIdx.x] = 0.f; }
    __syncthreads();
    int o = threadIdx.x & 63;
    int p = threadIdx.x >> 6;
    int point = blockIdx.x * 4 + p;
    int b = point >> 12;
    float a1 = P[(size_t)point * 128 + o];          // p1 = central projection
    const int* e = edges + (size_t)point * K_;
    __builtin_prefetch(e + K_, 0, 1);               // next point's edge list
    const float* P2 = P + (((size_t)b << 12) * 128) + 64;   // p2 block of this batch
    float s = 0.f, q = 0.f;
#pragma unroll
    for (int kk = 0; kk < K_; ++kk) {
        int m = e[kk];
        float h = a1 + P2[(size_t)m * 128 + o];
        s += h; q += h * h;
    }
    atomicAdd(&ssum[o], s);
    atomicAdd(&ssq2[o], q);
    __syncthreads();
    if (threadIdx.x < CO_) {
        atomicAdd(&gsum[threadIdx.x], ssum[threadIdx.x]);
        atomicAdd(&gsq[threadIdx.x],  ssq2[threadIdx.x]);
    }
}

// ---------------------------------------------------------------- BN scale/shift
__global__ void finalize_kernel(const float* __restrict__ gsum, const float* __restrict__ gsq,
                                const float* __restrict__ gamma, const float* __restrict__ beta,
                                float* __restrict__ scale, float* __restrict__ shift) {
    int o = threadIdx.x;
    if (o >= CO_) return;
    const float cnt = (float)((size_t)B_ * N_ * K_);
    float mean = gsum[o] / cnt;
    float var  = gsq[o] / cnt - mean * mean;
    float sc   = gamma[o] * rsqrtf(var + BN_EPS);
    scale[o] = sc;
    shift[o] = beta[o] - mean * sc;
}

// ---------------------------------------------------------------- BN + LeakyReLU + max_k, transposed store
__global__ void edge_out_kernel(const float* __restrict__ P, const int* __restrict__ edges,
                                const float* __restrict__ scale, const float* __restrict__ shift,
                                float* __restrict__ out) {
    __shared__ float tile[64][65];                  // 64 points x 64 channels (+pad)
    int o  = threadIdx.x & 63;
    int pb = threadIdx.x >> 6;
    int n0 = blockIdx.x * 64;                       // 64 consecutive points, same batch
    int b  = n0 >> 12;
    const float* P2 = P + (((size_t)b << 12) * 128) + 64;
    float sc = scale[o], sh = shift[o];

    for (int p = pb; p < 64; p += 4) {
        int point = n0 + p;
        float a1 = P[(size_t)point * 128 + o];
        const int* e = edges + (size_t)point * K_;
        __builtin_prefetch(edges + (size_t)(point + 4) * K_, 0, 1);
        float m = -3.0e38f;
#pragma unroll
        for (int kk = 0; kk < K_; ++kk) {
            int mm = e[kk];
            float h  = a1 + P2[(size_t)mm * 128 + o];
            float hn = h * sc + sh;
            float act = hn >= 0.f ? hn : SLOPE * hn;
            m = fmaxf(m, act);
        }
        tile[p][o] = m;
    }
    __syncthreads();
    int nloc = threadIdx.x & 63;
    for (int idx = threadIdx.x; idx < 64 * 64; idx += 256) {
        int o2 = idx >> 6;                          // coalesced: consecutive threads -> consecutive n
        out[(size_t)b * CO_ * N_ + (size_t)o2 * N_ + (n0 & (N_ - 1)) + nloc] = tile[nloc][o2];
    }
}

// ----------------------------------------------------------------
extern "C" void kernel_launch(void* const* d_in, const int* in_sizes, int n_in,
                              void* d_out, int out_size, void* d_ws, size_t ws_size,
                              hipStream_t stream) {
    (void)in_sizes; (void)n_in; (void)out_size; (void)ws_size;
    const float* x     = (const float*)d_in[0];
    const float* W     = (const float*)d_in[1];
    const float* gamma = (const float*)d_in[2];
    const float* beta  = (const float*)d_in[3];
    float* out = (float*)d_out;

    char* ws = (char*)d_ws;
    size_t off = 0;
    int*   edges = (int*)(ws + off);   off += (size_t)B_ * N_ * K_ * 4;   // 2.62 MB
    float* sq    = (float*)(ws + off); off += (size_t)B_ * N_ * 4;        // 0.13 MB
    float* wcatT = (float*)(ws + off); off += (size_t)C_ * 128 * 4;       // 32 KB
    float* P     = (float*)(ws + off); off += (size_t)B_ * N_ * 128 * 4;  // 16.8 MB (L2-resident)
    float* gsum  = (float*)(ws + off); off += CO_ * 4;
    float* gsq   = (float*)(ws + off); off += CO_ * 4;
    float* scale = (float*)(ws + off); off += CO_ * 4;
    float* shift = (float*)(ws + off); off += CO_ * 4;

    sq_kernel        <<<(B_ * N_) / 256, 256, 0, stream>>>(x, sq);
    knn_kernel       <<<B_ * (N_ / 256), 256, 0, stream>>>(x, sq, edges);
    wcat_kernel      <<<(C_ * 128) / 256, 256, 0, stream>>>(W, wcatT);
    proj_gemm_kernel <<<(B_ * N_) / GROWS, 256, 0, stream>>>(x, wcatT, P);
    init_stats_kernel<<<1, 128, 0, stream>>>(gsum);           // zeros gsum+gsq (contiguous)
    edge_stats_kernel<<<(B_ * N_) / 4, 256, 0, stream>>>(P, edges, gsum, gsq);
    finalize_kernel  <<<1, 64, 0, stream>>>(gsum, gsq, gamma, beta, scale, shift);
    edge_out_kernel  <<<(B_ * N_) / 64, 256, 0, stream>>>(P, edges, scale, shift, out);
}